// ModelNew_25056839204946
// MI455X (gfx1250) — compile-verified
//
#include <hip/hip_runtime.h>

typedef __attribute__((ext_vector_type(2))) float v2f;
typedef __attribute__((ext_vector_type(8))) float v8f;

#define SM_SCALE 0.17677669529663687f
#define SM_EPS   1e-12f

#define HW   4096
#define KK   9
#define DD   32
#define ROWS (DD * KK)   // 288 output rows per (b,h)
#define LPAD 33          // padded LDS row stride (dwords) -> conflict-free scatter
#define HW_TILE 32       // hw values per block

__global__ __launch_bounds__(256) void attn9_softmax_pv_wmma(
    const float* __restrict__ logits,   // [BH, HW, 9, 9]
    const float* __restrict__ vmat,     // [BH, HW, 9, 32]
    float* __restrict__ out)            // [B, H*288, HW]
{
    __shared__ float smem[ROWS * LPAD]; // 288*33*4 = 38016 B

    const int tid  = threadIdx.x;
    const int wave = tid >> 5;
    const int lane = tid & 31;
    const int half = lane >> 4;   // which 16-lane half
    const int mn   = lane & 15;   // A-row m / B,C-col n

    const int hwBlocks = HW / HW_TILE;              // 128
    const int bh       = blockIdx.x / hwBlocks;     // b*8 + h
    const int hw_base  = (blockIdx.x % hwBlocks) * HW_TILE;

    const float* logits_bh = logits + (size_t)bh * HW * (KK * KK);
    const float* v_bh      = vmat   + (size_t)bh * HW * (KK * DD);

    for (int it = 0; it < 4; ++it) {
        const int hwl = wave * 4 + it;          // 0..31 within tile
        const int hw  = hw_base + hwl;

        const float* lrow = logits_bh + (size_t)hw * (KK * KK) + mn * KK;
        const float* vhw  = v_bh      + (size_t)hw * (KK * DD);

        if (it < 3) {
            __builtin_prefetch(v_bh + (size_t)(hw + 1) * (KK * DD), 0, 1);
            __builtin_prefetch(logits_bh + (size_t)(hw + 1) * (KK * KK), 0, 1);
        }

        // ---- exact fp32 softmax of row m (duplicated across the two halves)
        float p[KK];
        if (mn < KK) {
            float x[KK];
            float mx = -3.402823466e+38f;
#pragma unroll
            for (int j = 0; j < KK; ++j) { x[j] = lrow[j] * SM_SCALE; mx = fmaxf(mx, x[j]); }
            float s = 0.f;
#pragma unroll
            for (int j = 0; j < KK; ++j) { p[j] = __expf(x[j] - mx); s += p[j]; }
            const float inv = 1.0f / (s + SM_EPS);
#pragma unroll
            for (int j = 0; j < KK; ++j) p[j] *= inv;
        } else {
#pragma unroll
            for (int j = 0; j < KK; ++j) p[j] = 0.f;
        }

        // ---- A fragments (16x4 f32): v0 = A[m][kb+2*half], v1 = A[m][kb+2*half+1]
        const int k0 = 2 * half;
        v2f a0, a1, a2;
        a0.x = p[k0];     a0.y = p[k0 + 1];
        a1.x = p[4 + k0]; a1.y = p[4 + k0 + 1];
        a2.x = (half == 0) ? p[8] : 0.f;   // k=8 valid only; k=9..11 are zero pad
        a2.y = 0.f;

        // ---- three K=4 chunks x two 16-wide d tiles
        v8f acc0 = {0.f, 0.f, 0.f, 0.f, 0.f, 0.f, 0.f, 0.f};
        v8f acc1 = {0.f, 0.f, 0.f, 0.f, 0.f, 0.f, 0.f, 0.f};

#pragma unroll
        for (int c = 0; c < 3; ++c) {
            const int kb = 4 * c + 2 * half;
            v2f b0, b1;
            b0.x = (kb     < KK) ? vhw[(kb    ) * DD + mn     ] : 0.f;
            b0.y = (kb + 1 < KK) ? vhw[(kb + 1) * DD + mn     ] : 0.f;
            b1.x = (kb     < KK) ? vhw[(kb    ) * DD + 16 + mn] : 0.f;
            b1.y = (kb + 1 < KK) ? vhw[(kb + 1) * DD + 16 + mn] : 0.f;

            const v2f a = (c == 0) ? a0 : ((c == 1) ? a1 : a2);

            acc0 = __builtin_amdgcn_wmma_f32_16x16x4_f32(
                       false, a, false, b0, (short)0, acc0, false, false);
            acc1 = __builtin_amdgcn_wmma_f32_16x16x4_f32(
                       false, a, false, b1, (short)0, acc1, false, false);
        }

        // ---- scatter C tiles into LDS: smem[(d*9 + i)*LPAD + hwl]
        // lane holds C[i = r + 8*half][d = t*16 + mn]; valid rows i<9
#pragma unroll
        for (int r = 0; r < 8; ++r) {
            const int i = r + 8 * half;
            if (i < KK) {
                smem[((0 * 16 + mn) * KK + i) * LPAD + hwl] = acc0[r];
                smem[((1 * 16 + mn) * KK + i) * LPAD + hwl] = acc1[r];
            }
        }
    }

    __syncthreads();

    // ---- coalesced drain: out[b, h*288 + row, hw_base + lane]
    const int b = bh >> 3;
    const int h = bh & 7;
    float* obase = out + (size_t)b * (8 * ROWS * HW) + (size_t)h * (ROWS * HW) + hw_base;
    for (int row = wave; row < ROWS; row += 8) {
        obase[(size_t)row * HW + lane] = smem[row * LPAD + lane];
    }
}

extern "C" void kernel_launch(void* const* d_in, const int* in_sizes, int n_in,
                              void* d_out, int out_size, void* d_ws, size_t ws_size,
                              hipStream_t stream) {
    const float* logits = (const float*)d_in[0];
    const float* vmat   = (const float*)d_in[1];
    float*       out    = (float*)d_out;

    const int bh_count = in_sizes[0] / (HW * KK * KK);     // B*H = 32
    const int grid     = bh_count * (HW / HW_TILE);        // 32 * 128 = 4096

    attn9_softmax_pv_wmma<<<grid, 256, 0, stream>>>(logits, vmat, out);
}